// ForwardWarpStereo_45208825757817
// MI455X (gfx1250) — compile-verified
//
#include <hip/hip_runtime.h>

#define W_   1280
#define H_   720
#define TPB  256
#define PER  (W_ / TPB)   // 5 pixels per thread

// ---------------------------------------------------------------------------
// Kernel 0: deterministic init of the global-min cell in workspace (+inf bits)
// ---------------------------------------------------------------------------
__global__ void fws_init_min(unsigned* gmin) {
    if (threadIdx.x == 0 && blockIdx.x == 0) *gmin = 0x7f800000u;  // +inf
}

// ---------------------------------------------------------------------------
// Kernel 1: global min of disp.  disp >= 0 so uint-bit ordering == float order.
// ---------------------------------------------------------------------------
__global__ __launch_bounds__(TPB)
void fws_min_reduce(const float* __restrict__ disp, unsigned* __restrict__ gmin,
                    long long n) {
    __shared__ float s[TPB];
    float m = __builtin_inff();
    for (long long i = (long long)blockIdx.x * TPB + threadIdx.x; i < n;
         i += (long long)gridDim.x * TPB)
        m = fminf(m, disp[i]);
    s[threadIdx.x] = m;
    __syncthreads();
    for (int off = TPB / 2; off > 0; off >>= 1) {
        if (threadIdx.x < off) s[threadIdx.x] = fminf(s[threadIdx.x], s[threadIdx.x + off]);
        __syncthreads();
    }
    if (threadIdx.x == 0) atomicMin(gmin, __float_as_uint(s[0]));
}

// ---------------------------------------------------------------------------
// Kernel 2: one workgroup per image row. Horizontal bilinear forward splat
// accumulated entirely in LDS (ds_add_f32), disp row staged via async
// global->LDS copy (ASYNCcnt path), then normalize + coalesced store.
// ---------------------------------------------------------------------------
__global__ __launch_bounds__(TPB)
void fws_warp_row(const float* __restrict__ im, const float* __restrict__ disp,
                  const float* __restrict__ dminp, float* __restrict__ out) {
    __shared__ float sDisp[W_];
    __shared__ float acc[4][W_];   // r,g,b accum + weight accum

    const int row = blockIdx.x;          // row = b*H + y
    const int b   = row / H_;
    const int y   = row - b * H_;
    const int tid = threadIdx.x;
    const float dmin = *dminp;           // bits written by atomicMin are a positive float

    // --- async-stage the disp row into LDS (gfx1250 ASYNCcnt path) ---
    const float* drow = disp + (size_t)row * W_;
    #pragma unroll
    for (int k = 0; k < PER; ++k) {
        const int x = tid + k * TPB;
        unsigned lds_off = (unsigned)(size_t)&sDisp[x];   // low 32 bits of shared addr = LDS byte offset
        const float* ga  = drow + x;
        asm volatile("global_load_async_to_lds_b32 %0, %1, off"
                     :: "v"(lds_off), "v"(ga) : "memory");
    }

    // --- zero the LDS accumulators while the async copy is in flight ---
    #pragma unroll
    for (int k = 0; k < PER; ++k) {
        const int x = tid + k * TPB;
        acc[0][x] = 0.f; acc[1][x] = 0.f; acc[2][x] = 0.f; acc[3][x] = 0.f;
    }
    asm volatile("s_wait_asynccnt 0" ::: "memory");
    __syncthreads();

    const size_t plane = (size_t)H_ * W_;
    const float* r0 = im + ((size_t)b * 3 + 0) * plane + (size_t)y * W_;
    const float* g0 = r0 + plane;
    const float* b0 = g0 + plane;

    #pragma unroll
    for (int k = 0; k < PER; ++k) {
        const int   x   = tid + k * TPB;
        const float d   = sDisp[x];
        // 1.414^(d-dmin) = exp2((d-dmin)*log2(1.414)), log2(1.414)=0.4997821
        const float wm  = __builtin_exp2f((d - dmin) * 0.4997821f);
        const float tx  = (float)x - d;             // flow_x = -disp, flow_y = 0
        const float x0f = floorf(tx);
        const int   x0  = (int)x0f;
        const int   x1  = x0 + 1;
        const float wc  = tx - x0f;                 // weight for x1
        const float wa  = 1.0f - wc;                // weight for x0
        const float r  = r0[x] * wm;
        const float g  = g0[x] * wm;
        const float bl = b0[x] * wm;
        if (x0 >= 0 && x0 < W_) {                   // ds_add_f32 (no return)
            atomicAdd(&acc[0][x0], r  * wa);
            atomicAdd(&acc[1][x0], g  * wa);
            atomicAdd(&acc[2][x0], bl * wa);
            atomicAdd(&acc[3][x0], wm * wa);
        }
        if (x1 >= 0 && x1 < W_) {
            atomicAdd(&acc[0][x1], r  * wc);
            atomicAdd(&acc[1][x1], g  * wc);
            atomicAdd(&acc[2][x1], bl * wc);
            atomicAdd(&acc[3][x1], wm * wc);
        }
    }
    __syncthreads();

    // --- normalize + coalesced store ---
    float* orow = out + ((size_t)b * 3) * plane + (size_t)y * W_;
    #pragma unroll
    for (int k = 0; k < PER; ++k) {
        const int   x   = tid + k * TPB;
        const float inv = 1.0f / fmaxf(acc[3][x], 1e-6f);
        orow[x]             = acc[0][x] * inv;
        orow[plane + x]     = acc[1][x] * inv;
        orow[2 * plane + x] = acc[2][x] * inv;
    }
}

// ---------------------------------------------------------------------------
extern "C" void kernel_launch(void* const* d_in, const int* in_sizes, int n_in,
                              void* d_out, int out_size, void* d_ws, size_t ws_size,
                              hipStream_t stream) {
    const float* im   = (const float*)d_in[0];   // [B,3,720,1280] fp32
    const float* disp = (const float*)d_in[1];   // [B,1,720,1280] fp32
    float*       out  = (float*)d_out;           // [B,3,720,1280] fp32
    unsigned*    gmin = (unsigned*)d_ws;

    const int       B = in_sizes[1] / (H_ * W_);
    const long long n = (long long)in_sizes[1];

    fws_init_min<<<1, 64, 0, stream>>>(gmin);
    fws_min_reduce<<<1024, TPB, 0, stream>>>(disp, gmin, n);
    fws_warp_row<<<B * H_, TPB, 0, stream>>>(im, disp, (const float*)gmin, out);
}